// multi_layer_GCN_3831110828045
// MI455X (gfx1250) — compile-verified
//
#include <hip/hip_runtime.h>
#include <hip/hip_bf16.h>

// ---------------------------------------------------------------------------
// Variational-GCN forward, fp32 WMMA (V_WMMA_F32_16X16X4_F32) on gfx1250.
// HBM-bound on streaming the 1 GiB fp32 adjacency twice (92 us floor at
// 23.3 TB/s). Double-buffered LDS pipeline:
//   - A (adj) tile: global_load_async_to_lds_b128 (ASYNCcnt-tracked DMA)
//   - B tile:       VGPR-staged, K-pair-interleaved in LDS so every WMMA B
//                   fragment is a single contiguous ds_load_b64 (no repacks)
// ---------------------------------------------------------------------------

typedef __attribute__((ext_vector_type(2))) float v2f;
typedef __attribute__((ext_vector_type(4))) float f32x4;
typedef __attribute__((ext_vector_type(8))) float v8f;

#define GCN_N 16384

__device__ __forceinline__ void async_ld_b128(void* lds_dst, const float* gsrc) {
  // VDST = LDS byte address (low 32 bits of generic shared pointer),
  // VADDR = 64-bit global address, saddr = off.
  unsigned lds_off = (unsigned)(size_t)lds_dst;
  unsigned long long ga = (unsigned long long)(size_t)gsrc;
  asm volatile("global_load_async_to_lds_b128 %0, %1, off"
               :: "v"(lds_off), "v"(ga)
               : "memory");
}

__device__ __forceinline__ void async_wait0() {
  asm volatile("s_wait_asynccnt 0x0" ::: "memory");
}

// ---------------------------------------------------------------------------
// C[M x F] = epilogue( A[M x K] @ B[K x F] ), 256 threads = 8 wave32 (4x2),
// block tile 64 x F, KT=32 K-slice, double-buffered LDS.
// WMMA f32 16x16x4 operand layout (ISA 7.12.2):
//   A: lane<16 -> row=lane, K+{0,1}; lane>=16 -> K+{2,3}  (contig float2)
//   B: VGPR0 = B[K+0|K+2][n], VGPR1 = B[K+1|K+3][n]       (pair-interleaved)
//   C/D: VGPR j -> (M = j + 8*(lane>>4), N = lane&15)
// ---------------------------------------------------------------------------
template <int F, int EPI>
__global__ __launch_bounds__(256) void gemm_wmma_f32(
    const float* __restrict__ A, const float* __restrict__ B,
    float* __restrict__ C, int M, int K) {
  constexpr int MT = 64;
  constexpr int KT = 32;
  constexpr int AKP = 36;           // padded A row stride (bank-conflict-free, 16B-aligned)
  constexpr int NTW = F / 32;       // n-tiles per wave
  constexpr int UNITS = F / 64;     // B staging units per thread (8 elems each)

  __shared__ __align__(16) float aS[2][MT * AKP];
  __shared__ __align__(16) float bS[2][KT * F];   // [k/2][n][2] pair-interleaved

  const int tid   = threadIdx.x;
  const int wid   = tid >> 5;
  const int lane  = tid & 31;
  const int lhalf = lane >> 4;
  const int lm    = lane & 15;
  const int mW    = (wid >> 1) * 16;
  const int nW    = (wid & 1) * (F / 2);
  const int mBlk  = blockIdx.x * MT;

  v8f acc[NTW];
  const v8f vzero = {0.f, 0.f, 0.f, 0.f, 0.f, 0.f, 0.f, 0.f};
#pragma unroll
  for (int t = 0; t < NTW; ++t) acc[t] = vzero;

  f32x4 rb0[UNITS], rb1[UNITS];

  // ---- staging helpers -----------------------------------------------------
  auto stageA_async = [&](int buf, int k0) {
#pragma unroll
    for (int i = 0; i < 2; ++i) {               // 64x32 f32 = 512 b128 chunks
      const int idx = tid + i * 256;
      const int r = idx >> 3, c4 = idx & 7;
      async_ld_b128(&aS[buf][r * AKP + c4 * 4],
                    A + (size_t)(mBlk + r) * K + k0 + c4 * 4);
    }
  };
  auto loadB = [&](int k0) {
#pragma unroll
    for (int uu = 0; uu < UNITS; ++uu) {
      const int u = uu * 256 + tid;
      const int kp = u / (F / 4), c4 = u % (F / 4);
      rb0[uu] = *(const f32x4*)(B + (size_t)(k0 + 2 * kp + 0) * F + 4 * c4);
      rb1[uu] = *(const f32x4*)(B + (size_t)(k0 + 2 * kp + 1) * F + 4 * c4);
    }
  };
  auto storeB = [&](int buf) {
#pragma unroll
    for (int uu = 0; uu < UNITS; ++uu) {
      const int u = uu * 256 + tid;
      const int kp = u / (F / 4), c4 = u % (F / 4);
      float* d = &bS[buf][kp * 2 * F + 8 * c4];
      const f32x4 lo = {rb0[uu].x, rb1[uu].x, rb0[uu].y, rb1[uu].y};
      const f32x4 hi = {rb0[uu].z, rb1[uu].z, rb0[uu].w, rb1[uu].w};
      ((f32x4*)d)[0] = lo;
      ((f32x4*)d)[1] = hi;
    }
  };

  // ---- prologue: fill buffer 0 --------------------------------------------
  stageA_async(0, 0);
  loadB(0);
  storeB(0);
  async_wait0();
  __syncthreads();

  const int nsteps = K / KT;
  for (int s = 0; s < nsteps; ++s) {
    const int k0 = s * KT;
    const int cur = s & 1, nxt = cur ^ 1;
    const bool more = (s + 1) < nsteps;
    if (more) {                      // launch next tile's traffic under compute
      stageA_async(nxt, k0 + KT);
      loadB(k0 + KT);
    }

    const float* aSc = aS[cur];
    const float* bSc = bS[cur];
#pragma unroll
    for (int kk = 0; kk < KT; kk += 4) {
      const v2f aF = *(const v2f*)(aSc + (mW + lm) * AKP + kk + 2 * lhalf);
#pragma unroll
      for (int t = 0; t < NTW; ++t) {
        const v2f bF =
            *(const v2f*)(bSc + ((kk >> 1) + lhalf) * 2 * F + 2 * (nW + t * 16 + lm));
        acc[t] = __builtin_amdgcn_wmma_f32_16x16x4_f32(
            false, aF, false, bF, (short)0, acc[t], false, false);
      }
    }

    if (more) {
      storeB(nxt);
      async_wait0();
    }
    __syncthreads();
  }

  // ---- epilogue + store ----------------------------------------------------
#pragma unroll
  for (int t = 0; t < NTW; ++t) {
    const int n0 = nW + t * 16;
#pragma unroll
    for (int j = 0; j < 8; ++j) {
      float v = acc[t][j];
      if (EPI == 1) v = tanhf(v);
      const int row = mBlk + mW + j + 8 * lhalf;
      C[(size_t)row * F + n0 + lm] = v;
    }
  }
}

// Pack Wcat[256][128] = [W_mean (256x64) | W_std (256x64)]
__global__ __launch_bounds__(256) void pack_w_kernel(
    const float* __restrict__ Wm, const float* __restrict__ Ws,
    float* __restrict__ Wcat) {
  const int i = blockIdx.x * 256 + threadIdx.x;
  const int r = i >> 7, c = i & 127;
  Wcat[i] = (c < 64) ? Wm[r * 64 + c] : Ws[r * 64 + (c - 64)];
}

// P[N][128] = [m | s_pre] -> out = concat(z, m, s), each N*64
__global__ __launch_bounds__(256) void finalize_kernel(
    const float* __restrict__ P, const float* __restrict__ eps,
    float* __restrict__ out, int nlat) {
  const int i = blockIdx.x * 256 + threadIdx.x;
  const int r = i >> 6, c = i & 63;
  const float m  = P[r * 128 + c];
  const float sp = P[r * 128 + 64 + c];
  const float s  = fmaxf(sp, 0.f) + 1e-4f;
  out[i]            = eps[i] * s + m;   // z
  out[nlat + i]     = m;
  out[2 * nlat + i] = s;
}

extern "C" void kernel_launch(void* const* d_in, const int* in_sizes, int n_in,
                              void* d_out, int out_size, void* d_ws, size_t ws_size,
                              hipStream_t stream) {
  const float* adj = (const float*)d_in[0];
  const float* x   = (const float*)d_in[1];
  const float* W0  = (const float*)d_in[2];
  const float* Wm  = (const float*)d_in[3];
  const float* Ws  = (const float*)d_in[4];
  const float* eps = (const float*)d_in[5];
  float* out = (float*)d_out;

  const int N = GCN_N;
  char* ws = (char*)d_ws;
  float* T0   = (float*)(ws);                                // N*256 (reused as P: N*128)
  float* h    = (float*)(ws + (size_t)N * 256 * 4);          // N*256
  float* HB   = (float*)(ws + (size_t)N * 256 * 4 * 2);      // N*128
  float* Wcat = (float*)(ws + (size_t)N * 256 * 4 * 2 + (size_t)N * 128 * 4);

  const int blocks = N / 64;

  gemm_wmma_f32<256, 0><<<blocks, 256, 0, stream>>>(x, W0, T0, N, 512);
  gemm_wmma_f32<256, 1><<<blocks, 256, 0, stream>>>(adj, T0, h, N, N);
  pack_w_kernel<<<(256 * 128) / 256, 256, 0, stream>>>(Wm, Ws, Wcat);
  gemm_wmma_f32<128, 0><<<blocks, 256, 0, stream>>>(h, Wcat, HB, N, 256);
  gemm_wmma_f32<128, 0><<<blocks, 256, 0, stream>>>(adj, HB, T0, N, N);
  finalize_kernel<<<(N * 64) / 256, 256, 0, stream>>>(T0, eps, out, N * 64);
}